// PowerSpectrumFeatures_86088324481925
// MI455X (gfx1250) — compile-verified
//
#include <hip/hip_runtime.h>
#include <hip/hip_bf16.h>

typedef __attribute__((ext_vector_type(2))) float v2f;
typedef __attribute__((ext_vector_type(8))) float v8f;

#define NM 16            // total m channels (l=0..3)
#define NQ 32            // N_PSEUDO * N_MAX
#define C_STRIDE (NM*NQ) // 512 floats per atom
#define CUTOFF 5.0f
#define PI_F 3.14159265358979323846f
#define LDS_STRIDE 33    // 32 cols + 1 pad to avoid bank conflicts

// ---------------------------------------------------------------- zero c
__global__ void ps_zero_kernel(float* __restrict__ p, long n) {
    long idx = (long)blockIdx.x * blockDim.x + threadIdx.x;
    long stride = (long)gridDim.x * blockDim.x;
    for (long k = idx; k < n; k += stride) p[k] = 0.0f;
}

// ---------------------------------------------------------------- edge scatter
// One wave (32 lanes) per edge. Lane q in [0,32) owns channel q = p*8 + n.
__global__ void ps_edge_kernel(const float* __restrict__ pos,
                               const float* __restrict__ cells,
                               const int*   __restrict__ species,
                               const int*   __restrict__ eidx,    // [2,E] flat
                               const int*   __restrict__ shifts,  // [E,3]
                               const float* __restrict__ embed,   // [4,4]
                               const float* __restrict__ mu,      // [8]
                               const float* __restrict__ sigma,   // [1]
                               float* __restrict__ c,             // [N,16,32]
                               int E) {
    int lane = threadIdx.x & 31;
    int e = (int)(((long)blockIdx.x * blockDim.x + threadIdx.x) >> 5);
    if (e >= E) return;                       // wave-uniform

    int i = eidx[e];
    int j = eidx[E + e];
    float s0 = (float)shifts[e * 3 + 0];
    float s1 = (float)shifts[e * 3 + 1];
    float s2 = (float)shifts[e * 3 + 2];

    float rx = pos[j * 3 + 0] - pos[i * 3 + 0] + s0 * cells[0] + s1 * cells[3] + s2 * cells[6];
    float ry = pos[j * 3 + 1] - pos[i * 3 + 1] + s0 * cells[1] + s1 * cells[4] + s2 * cells[7];
    float rz = pos[j * 3 + 2] - pos[i * 3 + 2] + s0 * cells[2] + s1 * cells[5] + s2 * cells[8];

    float r2 = rx * rx + ry * ry + rz * rz + 1e-12f;
    float r  = sqrtf(r2);
    float ir = 1.0f / r;
    float x = rx * ir, y = ry * ir, z = rz * ir;
    float x2 = x * x, y2 = y * y, z2 = z * z;

    float fcut = (r < CUTOFF) ? 0.5f * (__cosf(PI_F * r * (1.0f / CUTOFF)) + 1.0f) : 0.0f;

    // lane-private channel: p = lane>>3, n = lane&7
    int pp = lane >> 3, nn = lane & 7;
    float sig = sigma[0];
    float inv2s2 = 1.0f / (2.0f * sig * sig);
    float d = r - mu[nn];
    float Rn = __expf(-d * d * inv2s2) * fcut;
    float b = embed[species[j] * 4 + pp] * Rn;

    float Y[16];
    Y[0]  = 0.28209479177387814f;
    Y[1]  = 0.4886025119029199f * y;
    Y[2]  = 0.4886025119029199f * z;
    Y[3]  = 0.4886025119029199f * x;
    Y[4]  = 1.0925484305920792f * x * y;
    Y[5]  = 1.0925484305920792f * y * z;
    Y[6]  = 0.31539156525252005f * (3.0f * z2 - 1.0f);
    Y[7]  = 1.0925484305920792f * x * z;
    Y[8]  = 0.5462742152960396f * (x2 - y2);
    Y[9]  = 0.5900435899266435f * y * (3.0f * x2 - y2);
    Y[10] = 2.890611442640554f * x * y * z;
    Y[11] = 0.4570457994644658f * y * (5.0f * z2 - 1.0f);
    Y[12] = 0.3731763325901154f * z * (5.0f * z2 - 3.0f);
    Y[13] = 0.4570457994644658f * x * (5.0f * z2 - 1.0f);
    Y[14] = 1.445305721320277f * z * (x2 - y2);
    Y[15] = 0.5900435899266435f * x * (x2 - 3.0f * y2);

    float* cb = c + (long)i * C_STRIDE + lane;   // lane-contiguous -> coalesced atomics
#pragma unroll
    for (int m = 0; m < NM; ++m) {
        __hip_atomic_fetch_add(cb + m * NQ, Y[m] * b,
                               __ATOMIC_RELAXED, __HIP_MEMORY_SCOPE_AGENT);
    }
}

// ---------------------------------------------------------------- per-atom Gram via WMMA
// One wave per atom. ps_l = cg_l * c_l^T c_l, c_l = (2l+1) x 32 rows of c[atom].
// D = A x B with A[q',k] = c[m0+k][tq*16+q'], B[k,r'] = c[m0+k][tr*16+r'].
// f32 WMMA fragment layout (16x16x4): lanes 0-15 hold K = k,k+1; lanes 16-31 hold K = k+2,k+3.
__global__ void ps_atom_kernel(const float* __restrict__ c,
                               float* __restrict__ out,
                               int N) {
    __shared__ float lds[8][NM * LDS_STRIDE];
    int lane = threadIdx.x & 31;
    int w    = threadIdx.x >> 5;
    int atom = blockIdx.x * 8 + w;
    if (atom >= N) return;                    // wave-uniform; EXEC stays all-ones

    float* L = lds[w];
    const float* cs = c + (long)atom * C_STRIDE;
#pragma unroll
    for (int t = 0; t < NM; ++t)              // row t, col lane
        L[t * LDS_STRIDE + lane] = cs[t * NQ + lane];
    asm volatile("s_wait_dscnt 0" ::: "memory");  // cross-lane LDS RAW within wave

    int half = lane >> 4;       // which K pair this lane holds
    int col  = lane & 15;
    long obase = (long)atom * 4096;
    const float cg[4] = {1.0f, 0.5773502691896258f, 0.4472135954999579f, 0.3779644730092272f};

#pragma unroll
    for (int l = 0; l <= 3; ++l) {
        int mbase = l * l;
        int K = 2 * l + 1;
#pragma unroll
        for (int tq = 0; tq < 2; ++tq) {
#pragma unroll
            for (int tr = 0; tr < 2; ++tr) {
                v8f acc = {};
#pragma unroll
                for (int kc = 0; kc < K; kc += 4) {
                    int k0 = kc + 2 * half;
                    int k1 = k0 + 1;
                    float msk0 = (k0 < K) ? 1.0f : 0.0f;   // cndmask, no branch
                    float msk1 = (k1 < K) ? 1.0f : 0.0f;
                    int r0 = mbase + ((k0 < K) ? k0 : 0);
                    int r1 = mbase + ((k1 < K) ? k1 : 0);
                    v2f a, b;
                    a.x = L[r0 * LDS_STRIDE + tq * 16 + col] * msk0;  // masked A kills pad
                    a.y = L[r1 * LDS_STRIDE + tq * 16 + col] * msk1;
                    b.x = L[r0 * LDS_STRIDE + tr * 16 + col];
                    b.y = L[r1 * LDS_STRIDE + tr * 16 + col];
                    acc = __builtin_amdgcn_wmma_f32_16x16x4_f32(
                        false, a, false, b, (short)0, acc, false, false);
                }
                float s = cg[l];
#pragma unroll
                for (int v = 0; v < 8; ++v) {
                    int q = tq * 16 + v + 8 * half;   // D row for this VGPR/lane
                    int r = tr * 16 + col;            // D col
                    out[obase + l * 1024 + q * 32 + r] = s * acc[v];
                }
            }
        }
    }
}

// ---------------------------------------------------------------- launch
extern "C" void kernel_launch(void* const* d_in, const int* in_sizes, int n_in,
                              void* d_out, int out_size, void* d_ws, size_t ws_size,
                              hipStream_t stream) {
    const float* pos     = (const float*)d_in[0];
    const float* cells   = (const float*)d_in[1];
    const int*   species = (const int*)  d_in[2];
    const int*   eidx    = (const int*)  d_in[3];
    const int*   shifts  = (const int*)  d_in[4];
    const float* embed   = (const float*)d_in[5];
    const float* mu      = (const float*)d_in[6];
    const float* sigma   = (const float*)d_in[7];

    int N = in_sizes[0] / 3;
    int E = in_sizes[3] / 2;
    float* c = (float*)d_ws;                 // N*512 floats = 41 MB (fits L2)

    long cn = (long)N * C_STRIDE;
    ps_zero_kernel<<<1024, 256, 0, stream>>>(c, cn);

    int eblocks = (E + 7) / 8;               // 8 waves/block, 1 edge/wave
    ps_edge_kernel<<<eblocks, 256, 0, stream>>>(pos, cells, species, eidx, shifts,
                                                embed, mu, sigma, c, E);

    int ablocks = (N + 7) / 8;               // 8 waves/block, 1 atom/wave
    ps_atom_kernel<<<ablocks, 256, 0, stream>>>(c, (float*)d_out, N);
}